// SimilarityMatrix_63333587746928
// MI455X (gfx1250) — compile-verified
//
#include <hip/hip_runtime.h>
#include <math.h>

typedef __attribute__((ext_vector_type(2))) float v2f;
typedef __attribute__((ext_vector_type(8))) float v8f;

#define EPS 1e-6f
#define B_DIM 512
#define L_DIM 512
#define D_DIM 256
#define TILE 128     // output tile per workgroup (128x128)
#define KB 32        // K-chunk staged in LDS
#define KPAD 36      // padded LDS row stride (floats): 144B, 16B-aligned, bank-conflict-free

__global__ __launch_bounds__(256)
void pairwise_dist_wmma_f32(const float* __restrict__ s1,
                            const float* __restrict__ s2,
                            float* __restrict__ out) {
    __shared__ __align__(16) float lA[TILE * KPAD];
    __shared__ __align__(16) float lB[TILE * KPAD];
    __shared__ float sqA_s[TILE];
    __shared__ float sqB_s[TILE];

    const int tid  = threadIdx.x;
    const int lane = tid & 31;
    const int wave = tid >> 5;
    const int lm   = lane & 15;   // m (A) / n (B) within 16x16 tile
    const int kh   = lane >> 4;   // K half-select

    const int tileRow = (int)blockIdx.x >> 2;   // 0..3
    const int tileCol = (int)blockIdx.x & 3;    // 0..3
    const int b       = (int)blockIdx.y;        // batch

    // wave -> 32x64 sub-tile of the 128x128 output tile
    const int wr = (wave >> 1) * 32;  // row offset: 0,32,64,96
    const int wc = (wave & 1) * 64;   // col offset: 0,64

    const float* gAbase = s1 + (size_t)b * L_DIM * D_DIM + (size_t)(tileRow * TILE) * D_DIM;
    const float* gBbase = s2 + (size_t)b * L_DIM * D_DIM + (size_t)(tileCol * TILE) * D_DIM;

    v8f acc[2][4];
    #pragma unroll
    for (int tm = 0; tm < 2; ++tm)
        #pragma unroll
        for (int tn = 0; tn < 4; ++tn)
            acc[tm][tn] = (v8f){0.f, 0.f, 0.f, 0.f, 0.f, 0.f, 0.f, 0.f};

    float sqAcc = 0.0f;

    for (int k0 = 0; k0 < D_DIM; k0 += KB) {
        __syncthreads();
        // ---- cooperative global -> LDS: A[128][32], B[128][32] (B gets -eps) ----
        const float* gA = gAbase + k0;
        const float* gB = gBbase + k0;
        #pragma unroll
        for (int i = 0; i < 4; ++i) {
            int s = i * 256 + tid;          // 1024 float4 slots per matrix
            int r = s >> 3;                 // row 0..127
            int c = (s & 7) << 2;           // col 0,4,...,28
            float4 va = *(const float4*)(gA + (size_t)r * D_DIM + c);
            float4 vb = *(const float4*)(gB + (size_t)r * D_DIM + c);
            vb.x -= EPS; vb.y -= EPS; vb.z -= EPS; vb.w -= EPS;
            *(float4*)&lA[r * KPAD + c] = va;
            *(float4*)&lB[r * KPAD + c] = vb;
        }
        __syncthreads();

        // ---- accumulate row squared-norms from staged tiles ----
        {
            const float* lrow = (tid < 128) ? &lA[tid * KPAD] : &lB[(tid - 128) * KPAD];
            float s = 0.0f;
            #pragma unroll
            for (int c = 0; c < KB; ++c) s += lrow[c] * lrow[c];
            sqAcc += s;
        }

        // ---- WMMA over this K-chunk: 8 K-steps of 4 ----
        #pragma unroll
        for (int kk = 0; kk < 8; ++kk) {
            const int kb = kk * 4 + 2 * kh;
            v2f a0 = *(const v2f*)&lA[(wr +  0 + lm) * KPAD + kb];
            v2f a1 = *(const v2f*)&lA[(wr + 16 + lm) * KPAD + kb];
            v2f b0 = *(const v2f*)&lB[(wc +  0 + lm) * KPAD + kb];
            v2f b1 = *(const v2f*)&lB[(wc + 16 + lm) * KPAD + kb];
            v2f b2 = *(const v2f*)&lB[(wc + 32 + lm) * KPAD + kb];
            v2f b3 = *(const v2f*)&lB[(wc + 48 + lm) * KPAD + kb];

            acc[0][0] = __builtin_amdgcn_wmma_f32_16x16x4_f32(false, a0, false, b0, (short)0, acc[0][0], false, false);
            acc[0][1] = __builtin_amdgcn_wmma_f32_16x16x4_f32(false, a0, false, b1, (short)0, acc[0][1], false, false);
            acc[0][2] = __builtin_amdgcn_wmma_f32_16x16x4_f32(false, a0, false, b2, (short)0, acc[0][2], false, false);
            acc[0][3] = __builtin_amdgcn_wmma_f32_16x16x4_f32(false, a0, false, b3, (short)0, acc[0][3], false, false);
            acc[1][0] = __builtin_amdgcn_wmma_f32_16x16x4_f32(false, a1, false, b0, (short)0, acc[1][0], false, false);
            acc[1][1] = __builtin_amdgcn_wmma_f32_16x16x4_f32(false, a1, false, b1, (short)0, acc[1][1], false, false);
            acc[1][2] = __builtin_amdgcn_wmma_f32_16x16x4_f32(false, a1, false, b2, (short)0, acc[1][2], false, false);
            acc[1][3] = __builtin_amdgcn_wmma_f32_16x16x4_f32(false, a1, false, b3, (short)0, acc[1][3], false, false);
        }
    }

    // ---- publish row norms ----
    if (tid < 128) sqA_s[tid] = sqAcc;
    else           sqB_s[tid - 128] = sqAcc;
    __syncthreads();

    // ---- epilogue: d = sqrt(max(sqA + sqB - 2*cross, 0)) ----
    float* outB = out + (size_t)b * L_DIM * L_DIM;
    const int rowBase = tileRow * TILE;
    const int colBase = tileCol * TILE;
    #pragma unroll
    for (int tm = 0; tm < 2; ++tm) {
        #pragma unroll
        for (int tn = 0; tn < 4; ++tn) {
            const int cl  = wc + tn * 16 + lm;
            const float sqb = sqB_s[cl];
            #pragma unroll
            for (int v = 0; v < 8; ++v) {
                const int rl = wr + tm * 16 + v + 8 * kh;   // C/D layout: M = v + 8*(lane>>4)
                float d2 = sqA_s[rl] + sqb - 2.0f * acc[tm][tn][v];
                outB[(size_t)(rowBase + rl) * L_DIM + (colBase + cl)] = sqrtf(fmaxf(d2, 0.0f));
            }
        }
    }
}

extern "C" void kernel_launch(void* const* d_in, const int* in_sizes, int n_in,
                              void* d_out, int out_size, void* d_ws, size_t ws_size,
                              hipStream_t stream) {
    const float* s1 = (const float*)d_in[0];
    const float* s2 = (const float*)d_in[1];
    float* out = (float*)d_out;
    dim3 grid(16, B_DIM, 1);   // 4x4 tiles of 128x128 per batch
    dim3 block(256, 1, 1);     // 8 waves (wave32)
    pairwise_dist_wmma_f32<<<grid, block, 0, stream>>>(s1, s2, out);
}